// Uni3FC_68719477137
// MI455X (gfx1250) — compile-verified
//
#include <hip/hip_runtime.h>
#include <hip/hip_bf16.h>

typedef __attribute__((ext_vector_type(2))) float v2f;
typedef __attribute__((ext_vector_type(8))) float v8f;

#define Bn 4
#define Cc 64
#define Nn 8192
#define Kk 40
#define Hh 4
#define Dd 16
#define FFd 256
#define BNEPS 1e-5f
#define KP 68  // padded LDS row stride (floats): 68*4B = 272B, 16B aligned, bank-skew 4

__device__ __forceinline__ v8f wmma4(v2f a, v2f b, v8f c) {
  // D = A(16x4) * B(4x16) + C, fp32
  return __builtin_amdgcn_wmma_f32_16x16x4_f32(false, a, false, b, (short)0, c,
                                               false, false);
}

// ---- CDNA5 async Global->LDS DMA (ASYNCcnt-tracked) -----------------------
__device__ __forceinline__ unsigned lds_off(const void* p) {
  // generic -> AS3 -> 32-bit LDS byte offset
  return (unsigned)(uintptr_t)(__attribute__((address_space(3))) const void*)p;
}

__device__ __forceinline__ void async_load_b128(unsigned lds_byte, const void* gptr) {
  asm volatile("global_load_async_to_lds_b128 %0, %1, off"
               :
               : "v"(lds_byte), "v"((unsigned long long)(uintptr_t)gptr)
               : "memory");
}

template <int N>
__device__ __forceinline__ void wait_asynccnt() {
  asm volatile("s_wait_asynccnt %0" ::"i"(N) : "memory");
}

// ---------------- prep: transpose x (B,C,N) -> pcd (B,N,C); norms ----------
__global__ void k_transpose(const float* __restrict__ x, float* __restrict__ pcd) {
  int b = blockIdx.y, c = blockIdx.x;
  for (int n = threadIdx.x; n < Nn; n += blockDim.x)
    pcd[((size_t)b * Nn + n) * Cc + c] = x[((size_t)b * Cc + c) * Nn + n];
}

__global__ void k_norm(const float* __restrict__ pcd, float* __restrict__ aa) {
  int p = blockIdx.x * blockDim.x + threadIdx.x;
  if (p >= Bn * Nn) return;
  const float* r = pcd + (size_t)p * Cc;
  float s = 0.f;
#pragma unroll
  for (int c = 0; c < Cc; ++c) s += r[c] * r[c];
  aa[p] = s;
}

// ---------------- full-point projections q/k/v via WMMA --------------------
// out[n,o] = sum_c pcd[n,c] * W[o,c]   (rows = B*N points)
__global__ __launch_bounds__(128) void k_proj(const float* __restrict__ pcd,
                                              const float* __restrict__ Wq,
                                              const float* __restrict__ Wk,
                                              const float* __restrict__ Wv,
                                              float* __restrict__ qf,
                                              float* __restrict__ kf,
                                              float* __restrict__ vf) {
  const int row0 = blockIdx.x * 16;
  const int lane = threadIdx.x & 31;
  const int wave = threadIdx.x >> 5;  // output col tile 0..3
  const int rowm = lane & 15;
  const int hi = lane >> 4;

  v2f a[16];
  const float* ar = pcd + (size_t)(row0 + rowm) * Cc;
#pragma unroll
  for (int k = 0; k < 16; ++k) {
    int c0 = k * 4 + hi * 2;
    a[k][0] = ar[c0];
    a[k][1] = ar[c0 + 1];
  }

  const float* Ws[3] = {Wq, Wk, Wv};
  float* Os[3] = {qf, kf, vf};
#pragma unroll
  for (int m = 0; m < 3; ++m) {
    const float* W = Ws[m];
    v8f acc = {};
#pragma unroll
    for (int k = 0; k < 16; ++k) {
      int c0 = k * 4 + hi * 2;
      int o = wave * 16 + rowm;
      v2f bv;
      bv[0] = W[o * Cc + c0];
      bv[1] = W[o * Cc + c0 + 1];
      acc = wmma4(a[k], bv, acc);
    }
    float* O = Os[m];
#pragma unroll
    for (int r = 0; r < 8; ++r) {
      int M = r + 8 * hi;
      O[(size_t)(row0 + M) * Cc + wave * 16 + rowm] = acc[r];
    }
  }
}

// ---------------- kNN: async-staged keys + WMMA + fused top-40 -------------
__global__ __launch_bounds__(128) void k_knn(const float* __restrict__ pcd,
                                             const float* __restrict__ aa,
                                             int* __restrict__ idx_out) {
  __shared__ float s_keys[2][64][KP];  // double-buffered 64-key tile
  __shared__ float s_sc[16][64];
  __shared__ float s_topv[16][Kk];
  __shared__ int s_topi[16][Kk];
  __shared__ float s_minv[16];
  __shared__ int s_minp[16];
  __shared__ float s_aaq[16];

  const int b = blockIdx.y;
  const int q0 = blockIdx.x * 16;
  const int tid = threadIdx.x;
  const int lane = tid & 31;
  const int wave = tid >> 5;
  const int rowm = lane & 15;
  const int hi = lane >> 4;

  const float* pb = pcd + (size_t)b * Nn * Cc;
  const float* ab = aa + (size_t)b * Nn;

  // query A-operand (16 rows x C), resident in VGPRs for the whole sweep
  v2f aq[16];
  {
    const float* qr = pb + (size_t)(q0 + rowm) * Cc;
#pragma unroll
    for (int k = 0; k < 16; ++k) {
      int c0 = k * 4 + hi * 2;
      aq[k][0] = qr[c0];
      aq[k][1] = qr[c0 + 1];
    }
  }
  if (tid < 16) {
    s_aaq[tid] = ab[q0 + tid];
    s_minv[tid] = -3.0e38f;
    s_minp[tid] = 0;
    for (int t = 0; t < Kk; ++t) {
      s_topv[tid][t] = -3.0e38f;
      s_topi[tid][t] = 0;
    }
  }

  // stage key tile kb into s_keys[buf]: 64 rows x 16 b128 chunks = 1024 chunks
  // 128 threads x 8 chunks; per-wave ASYNCcnt += 8
  auto stage = [&](int buf, int kb) {
#pragma unroll
    for (int i = 0; i < 8; ++i) {
      int chunk = tid + i * 128;
      int r = chunk >> 4;
      int c4 = (chunk & 15) * 4;
      async_load_b128(lds_off(&s_keys[buf][r][c4]),
                      pb + (size_t)(kb + r) * Cc + c4);
    }
  };

  stage(0, 0);
  __syncthreads();

  int buf = 0;
  for (int kb = 0; kb < Nn; kb += 64) {
    if (kb + 64 < Nn) {
      stage(buf ^ 1, kb + 64);   // overlap next tile DMA with this tile's WMMA
      wait_asynccnt<8>();        // retire current tile (async loads in order)
    } else {
      wait_asynccnt<0>();
    }
    __syncthreads();  // all waves' stages of this tile complete

    const int key = kb + wave * 16 + rowm;
    const float* kr = &s_keys[buf][wave * 16 + rowm][0];
    v8f acc = {};
#pragma unroll
    for (int k = 0; k < 16; ++k) {
      int c0 = k * 4 + hi * 2;
      v2f bv;
      bv[0] = kr[c0];
      bv[1] = kr[c0 + 1];
      acc = wmma4(aq[k], bv, acc);
    }
    float aak = ab[key];
#pragma unroll
    for (int r = 0; r < 8; ++r) {
      int m = r + 8 * hi;
      s_sc[m][wave * 16 + rowm] = 2.0f * acc[r] - s_aaq[m] - aak;
    }
    __syncthreads();
    if (tid < 16) {  // per-query running top-40 (maximize pd = -dist^2)
      int row = tid;
      float mv = s_minv[row];
      int mp = s_minp[row];
      for (int c2 = 0; c2 < 64; ++c2) {
        float v = s_sc[row][c2];
        if (v > mv) {
          s_topv[row][mp] = v;
          s_topi[row][mp] = kb + c2;
          mv = s_topv[row][0];
          mp = 0;
          for (int t = 1; t < Kk; ++t) {
            float tv = s_topv[row][t];
            if (tv < mv) { mv = tv; mp = t; }
          }
        }
      }
      s_minv[row] = mv;
      s_minp[row] = mp;
    }
    __syncthreads();  // also fences s_keys[buf] reads before re-staging it
    buf ^= 1;
  }
  if (tid < 16) {
    int row = tid;
    int* o = idx_out + ((size_t)b * Nn + q0 + row) * Kk;
    for (int t = 0; t < Kk; ++t) o[t] = s_topi[row][t];
  }
}

// ---------------- attention: async-gather projected neighbors --------------
__global__ __launch_bounds__(128) void k_attn(const float* __restrict__ qf,
                                              const float* __restrict__ kf,
                                              const float* __restrict__ vf,
                                              const int* __restrict__ idx,
                                              float* __restrict__ tmp) {
  __shared__ float s_q[Cc], s_kc[Cc], s_vc[Cc];
  __shared__ int s_idx[Kk];
  __shared__ float s_a[Hh][Kk];
  __shared__ float s_kn[Kk][KP];  // gathered k rows (padded vs bank conflicts)

  const int b = blockIdx.y;
  const int n = blockIdx.x;
  const size_t base = (size_t)b * Nn;
  const int t = threadIdx.x;
  if (t < Cc) {
    s_q[t] = qf[(base + n) * Cc + t];
    s_kc[t] = kf[(base + n) * Cc + t];
    s_vc[t] = vf[(base + n) * Cc + t];
  }
  if (t < Kk) s_idx[t] = idx[(base + n) * Kk + t];
  __syncthreads();

  // async-gather 40 neighbor k-rows: 40*16 = 640 b128 chunks, 5 per thread
#pragma unroll
  for (int i = 0; i < 5; ++i) {
    int chunk = t + i * 128;
    int j = chunk >> 4;
    int c4 = (chunk & 15) * 4;
    async_load_b128(lds_off(&s_kn[j][c4]), kf + (base + s_idx[j]) * Cc + c4);
  }
  wait_asynccnt<0>();
  __syncthreads();

  for (int task = t; task < Hh * Kk; task += blockDim.x) {
    int h = task / Kk, j = task % Kk;
    float e = 0.f;
#pragma unroll
    for (int d = 0; d < Dd; ++d)
      e += s_q[h * Dd + d] * (s_kn[j][h * Dd + d] - s_kc[h * Dd + d]);
    s_a[h][j] = e * 0.25f;  // / sqrt(D), D = 16
  }
  __syncthreads();
  if (t < Hh) {
    float mx = -3.0e38f;
    for (int j = 0; j < Kk; ++j) mx = fmaxf(mx, s_a[t][j]);
    float sum = 0.f;
    for (int j = 0; j < Kk; ++j) {
      float ev = __expf(s_a[t][j] - mx);
      s_a[t][j] = ev;
      sum += ev;
    }
    float inv = 1.f / sum;
    for (int j = 0; j < Kk; ++j) s_a[t][j] *= inv;
  }
  __syncthreads();
  if (t < Cc) {
    int h = t >> 4;
    float acc = 0.f;
    for (int j = 0; j < Kk; ++j)
      acc += s_a[h][j] * (vf[(base + s_idx[j]) * Cc + t] - s_vc[t]);
    tmp[((size_t)b * Cc + t) * Nn + n] = acc;  // (B,C,N)
  }
}

// ---------------- BatchNorm: stats + apply (y = a (+ b2) ) ----------------
__global__ __launch_bounds__(256) void k_bn_stats(const float* __restrict__ a,
                                                  const float* __restrict__ b2,
                                                  float* __restrict__ mean,
                                                  float* __restrict__ var) {
  __shared__ float s1[256], s2[256];
  const int c = blockIdx.x;
  float sum = 0.f, sq = 0.f;
  for (int i = threadIdx.x; i < Bn * Nn; i += 256) {
    int bb = i / Nn, n = i - bb * Nn;
    size_t off = ((size_t)bb * Cc + c) * Nn + n;
    float v = a[off];
    if (b2) v += b2[off];
    sum += v;
    sq += v * v;
  }
  s1[threadIdx.x] = sum;
  s2[threadIdx.x] = sq;
  __syncthreads();
  for (int s = 128; s > 0; s >>= 1) {
    if (threadIdx.x < s) {
      s1[threadIdx.x] += s1[threadIdx.x + s];
      s2[threadIdx.x] += s2[threadIdx.x + s];
    }
    __syncthreads();
  }
  if (threadIdx.x == 0) {
    float m = s1[0] / (float)(Bn * Nn);
    mean[c] = m;
    var[c] = s2[0] / (float)(Bn * Nn) - m * m;  // biased
  }
}

__global__ void k_bn_apply(const float* __restrict__ a, const float* __restrict__ b2,
                           const float* __restrict__ mean, const float* __restrict__ var,
                           const float* __restrict__ g, const float* __restrict__ be,
                           float* __restrict__ out) {
  size_t i = (size_t)blockIdx.x * blockDim.x + threadIdx.x;
  if (i >= (size_t)Bn * Cc * Nn) return;
  int c = (int)((i / Nn) % Cc);
  float v = a[i];
  if (b2) v += b2[i];
  out[i] = (v - mean[c]) * rsqrtf(var[c] + BNEPS) * g[c] + be[c];
}

// ---------------- FFN1: h = leaky_relu(W1 @ x1), (B,FF,N) ------------------
__global__ __launch_bounds__(128) void k_ffn1(const float* __restrict__ W1,
                                              const float* __restrict__ x1,
                                              float* __restrict__ hbuf) {
  const int b = blockIdx.y;
  const int n0 = blockIdx.x * 16;
  const int lane = threadIdx.x & 31;
  const int wave = threadIdx.x >> 5;
  const int rowm = lane & 15, hi = lane >> 4;

  for (int ot = wave; ot < FFd / 16; ot += 4) {
    v8f acc = {};
#pragma unroll
    for (int k = 0; k < 16; ++k) {
      int c0 = k * 4 + hi * 2;
      int o = ot * 16 + rowm;
      v2f av, bv;
      av[0] = W1[o * Cc + c0];
      av[1] = W1[o * Cc + c0 + 1];
      bv[0] = x1[((size_t)b * Cc + c0) * Nn + n0 + rowm];
      bv[1] = x1[((size_t)b * Cc + c0 + 1) * Nn + n0 + rowm];
      acc = wmma4(av, bv, acc);
    }
#pragma unroll
    for (int r = 0; r < 8; ++r) {
      int M = r + 8 * hi;
      float vv = acc[r];
      hbuf[((size_t)b * FFd + ot * 16 + M) * Nn + n0 + rowm] =
          vv > 0.f ? vv : 0.2f * vv;
    }
  }
}

// ---------------- FFN2: z = x1 + W2 @ h, (B,C,N) ---------------------------
__global__ __launch_bounds__(128) void k_ffn2(const float* __restrict__ W2,
                                              const float* __restrict__ hbuf,
                                              const float* __restrict__ x1,
                                              float* __restrict__ z) {
  const int b = blockIdx.y;
  const int n0 = blockIdx.x * 16;
  const int lane = threadIdx.x & 31;
  const int wave = threadIdx.x >> 5;  // output channel tile 0..3
  const int rowm = lane & 15, hi = lane >> 4;
  v8f acc = {};
  for (int k = 0; k < FFd / 4; ++k) {
    int o0 = k * 4 + hi * 2;
    int crow = wave * 16 + rowm;
    v2f av, bv;
    av[0] = W2[crow * FFd + o0];
    av[1] = W2[crow * FFd + o0 + 1];
    bv[0] = hbuf[((size_t)b * FFd + o0) * Nn + n0 + rowm];
    bv[1] = hbuf[((size_t)b * FFd + o0 + 1) * Nn + n0 + rowm];
    acc = wmma4(av, bv, acc);
  }
#pragma unroll
  for (int r = 0; r < 8; ++r) {
    int M = r + 8 * hi;
    size_t off = ((size_t)b * Cc + wave * 16 + M) * Nn + n0 + rowm;
    z[off] = x1[off] + acc[r];
  }
}

extern "C" void kernel_launch(void* const* d_in, const int* in_sizes, int n_in,
                              void* d_out, int out_size, void* d_ws, size_t ws_size,
                              hipStream_t stream) {
  const float* x  = (const float*)d_in[0];
  const float* Wq = (const float*)d_in[1];
  const float* Wk = (const float*)d_in[2];
  const float* Wv = (const float*)d_in[3];
  const float* W1 = (const float*)d_in[4];
  const float* W2 = (const float*)d_in[5];
  const float* g1 = (const float*)d_in[6];
  const float* b1 = (const float*)d_in[7];
  const float* g2 = (const float*)d_in[8];
  const float* b2 = (const float*)d_in[9];

  float* ws = (float*)d_ws;
  const size_t E = (size_t)Bn * Cc * Nn;  // 2,097,152
  float* pcd = ws;
  float* aa  = ws + E;
  float* qf  = aa + (size_t)Bn * Nn;
  float* kf  = qf + E;
  float* vf  = kf + E;
  int*   idx = (int*)(vf + E);
  float* tmp = (float*)(idx + (size_t)Bn * Nn * Kk);
  float* x1  = tmp + E;
  float* st  = x1 + E;           // mean1,var1,mean2,var2 (4*64)
  float* hbuf = qf;              // alias: q/k/v/idx/tmp dead before FFN1
  float* z    = pcd;             // alias: pcd dead after attention

  k_transpose<<<dim3(Cc, Bn), 256, 0, stream>>>(x, pcd);
  k_norm<<<(Bn * Nn + 255) / 256, 256, 0, stream>>>(pcd, aa);
  k_proj<<<(Bn * Nn) / 16, 128, 0, stream>>>(pcd, Wq, Wk, Wv, qf, kf, vf);
  k_knn<<<dim3(Nn / 16, Bn), 128, 0, stream>>>(pcd, aa, idx);
  k_attn<<<dim3(Nn, Bn), 128, 0, stream>>>(qf, kf, vf, idx, tmp);
  k_bn_stats<<<Cc, 256, 0, stream>>>(x, tmp, st, st + 64);
  k_bn_apply<<<(int)((E + 255) / 256), 256, 0, stream>>>(x, tmp, st, st + 64, g1, b1, x1);
  k_ffn1<<<dim3(Nn / 16, Bn), 128, 0, stream>>>(W1, x1, hbuf);
  k_ffn2<<<dim3(Nn / 16, Bn), 128, 0, stream>>>(W2, hbuf, x1, z);
  k_bn_stats<<<Cc, 256, 0, stream>>>(z, nullptr, st + 128, st + 192);
  k_bn_apply<<<(int)((E + 255) / 256), 256, 0, stream>>>(z, nullptr, st + 128, st + 192,
                                                         g2, b2, (float*)d_out);
}